// OctantQueryMask_33414845563627
// MI455X (gfx1250) — compile-verified
//
#include <hip/hip_runtime.h>

typedef float v2f __attribute__((ext_vector_type(2)));
typedef float v4f __attribute__((ext_vector_type(4)));
typedef float v8f __attribute__((ext_vector_type(8)));

#define RADIUS2 0.04f   // 0.2^2
#define KMAX 16
#define WAVES_PER_BLOCK 4

// One wave handles 16 query rows [i0, i0+16).
// Distance tile via v_wmma_f32_16x16x4_f32:
//   A[i,:] = ( x_i,  y_i,  z_i, 1 )
//   B[:,j] = (-2x_j,-2y_j,-2z_j, |p_j|^2 )
//   C[i,:] = |p_i|^2          =>  D = dist^2 (16x16)
// D layout: VGPR r of lane L holds (row = r + 8*(L>=16), col = L%16).
// Selection: per row, 3 octant-bit ballots + a within ballot. All per-lane
// octant state is re-derived from the ballot words (1 v_cmp per bit), the
// same-octant group mask is pure bitop algebra, and rows with an empty
// within-ballot are skipped via a uniform branch (~34% of row steps).
// Per-(row,octant) counters live in LDS (wave-private slice, no barriers).
// Index slots are pre-filled with -1 (b128 stores); epilogue writes masks.
__global__ __launch_bounds__(32 * WAVES_PER_BLOCK)
void OctantQuery_wmma_kernel(const float* __restrict__ pcs,
                             float* __restrict__ out_idx,
                             float* __restrict__ out_msk,
                             int N) {
    __shared__ int cntLDS[WAVES_PER_BLOCK][16][8];

    const int tid  = threadIdx.x;
    const int wave = tid >> 5;
    const int lane = tid & 31;
    const int lj   = lane & 15;      // column within tile / slot id
    const int hi   = lane >> 4;      // half-wave select
    const unsigned half_mask = hi ? 0xFFFF0000u : 0x0000FFFFu;
    const unsigned lt_mask   = (1u << lane) - 1u;

    const int tiles_per_batch = N >> 4;
    const int tile = blockIdx.x * WAVES_PER_BLOCK + wave;
    const int b  = tile / tiles_per_batch;
    const int i0 = (tile % tiles_per_batch) << 4;

    const float* X = pcs + (size_t)b * 3 * N;
    const float* Y = X + N;
    const float* Z = Y + N;

    // Wave-uniform output bases for this wave's 16 rows (rows contiguous).
    float* idxBase = out_idx + ((size_t)b * N + i0) * 128;   // [16][8][16]
    float* mskBase = out_msk + ((size_t)b * N + i0) * 128;   // [16][8][16]

    // Pin the hot-loop store base into SGPRs (it is wave-uniform) so hit
    // stores can use the SADDR + 32-bit-voffset addressing form.
    float* idxBaseS;
    {
        unsigned long long p = (unsigned long long)idxBase;
        unsigned plo = __builtin_amdgcn_readfirstlane((unsigned)p);
        unsigned phi = __builtin_amdgcn_readfirstlane((unsigned)(p >> 32));
        idxBaseS = (float*)(((unsigned long long)phi << 32) | plo);
    }

    // Prologue 1: default-fill all index slots with -1 (coalesced b128).
    {
        const v4f neg1 = {-1.0f, -1.0f, -1.0f, -1.0f};
#pragma unroll
        for (int row = 0; row < 16; ++row)
            *(v4f*)(idxBase + row * 128 + lane * 4) = neg1;
    }

    int* cw = &cntLDS[wave][0][0];   // private slice, row-major [16][8]

    // Prologue 2: zero own counter slice (4 ints per lane). Same-wave LDS is
    // in-order and slices are wave-private, so no barrier is required.
#pragma unroll
    for (int t = 0; t < 4; ++t) cw[lane + 32 * t] = 0;

    // Query coords + squared norms for the 8 rows owned by this half-wave
    // (uniform across the 16 lanes of a half).
    float qx[8], qy[8], qz[8], csq[8];
#pragma unroll
    for (int r = 0; r < 8; ++r) {
        int i = i0 + 8 * hi + r;
        qx[r]  = X[i];
        qy[r]  = Y[i];
        qz[r]  = Z[i];
        csq[r] = qx[r] * qx[r] + qy[r] * qy[r] + qz[r] * qz[r];
    }

    // A operand: row i = i0 + lj, K-pair = (2*hi, 2*hi+1).
    // lanes 0-15 : (x_i, y_i)   lanes 16-31 : (z_i, 1)
    v2f a;
    {
        int ia = i0 + lj;
        float ax = X[ia], ay = Y[ia], az = Z[ia];
        a.x = hi ? az   : ax;
        a.y = hi ? 1.0f : ay;
    }

    for (int j0 = 0; j0 < N; j0 += 16) {
        const int j = j0 + lj;
        const float px = X[j], py = Y[j], pz = Z[j];
        const float sqj = px * px + py * py + pz * pz;

        // B operand: col j, K-pair = (2*hi, 2*hi+1).
        // lanes 0-15 : (-2x_j, -2y_j)   lanes 16-31 : (-2z_j, |p_j|^2)
        v2f bv;
        bv.x = hi ? (-2.0f * pz) : (-2.0f * px);
        bv.y = hi ? sqj          : (-2.0f * py);

        v8f c;
#pragma unroll
        for (int r = 0; r < 8; ++r) c[r] = csq[r];

        // D = A x B + C  ->  dist^2 tile
        v8f d = __builtin_amdgcn_wmma_f32_16x16x4_f32(
            /*neg_a=*/false, a, /*neg_b=*/false, bv,
            /*c_mod=*/(short)0, c, /*reuse_a=*/false, /*reuse_b=*/false);

        const float fj = (float)j;

#pragma unroll
        for (int r = 0; r < 8; ++r) {
            const bool within = d[r] < RADIUS2;
            const unsigned bw = __builtin_amdgcn_ballot_w32(within);

            // Wave-uniform skip: no lane (in either half) has a hit here.
            if (bw != 0u) {
                const unsigned b2 = __builtin_amdgcn_ballot_w32(px > qx[r]);
                const unsigned b1 = __builtin_amdgcn_ballot_w32(py > qy[r]);
                const unsigned b0 = __builtin_amdgcn_ballot_w32(pz > qz[r]);

                // My octant bits, re-derived from the ballot words.
                const unsigned gxb = (b2 >> lane) & 1u;
                const unsigned gyb = (b1 >> lane) & 1u;
                const unsigned gzb = (b0 >> lane) & 1u;

                // lanes in my half, within radius, same octant as me
                const unsigned same = bw
                    & ~(b2 ^ (0u - gxb))
                    & ~(b1 ^ (0u - gyb))
                    & ~(b0 ^ (0u - gzb))
                    & half_mask;

                const int pre   = __popc(same & lt_mask);  // rank in group
                const int total = __popc(same);            // group size
                const int oct   = (int)((gxb << 2) | (gyb << 1) | gzb);
                const int rowLocal = 8 * hi + r;
                int* cp = cw + rowLocal * 8 + oct;

                const int base = *cp;   // group-uniform broadcast read

                if (within && (base + pre) < KMAX) {
                    const unsigned off =
                        (unsigned)(rowLocal * 128 + oct * 16 + base + pre);
                    idxBaseS[off] = fj;
                }
                if (within && pre == 0)     // group leader bumps the counter
                    *cp = base + total;
            }
        }
    }

    // Epilogue: masks only (indices already defaulted in the prologue).
    // Whole wave covers one row per step: lane owns 4 consecutive slots,
    // all in octant (lane >> 2).
    const int octL = lane >> 2;          // my octant within the row
    const int s0   = (lane * 4) & 15;    // my first slot within that octant
#pragma unroll
    for (int row = 0; row < 16; ++row) {
        int cc = cw[row * 8 + octL];
        cc = (cc > KMAX) ? KMAX : cc;
        v4f m;
        m.x = (s0 + 0 < cc) ? 1.0f : 0.0f;
        m.y = (s0 + 1 < cc) ? 1.0f : 0.0f;
        m.z = (s0 + 2 < cc) ? 1.0f : 0.0f;
        m.w = (s0 + 3 < cc) ? 1.0f : 0.0f;
        *(v4f*)(mskBase + row * 128 + lane * 4) = m;
    }
}

extern "C" void kernel_launch(void* const* d_in, const int* in_sizes, int n_in,
                              void* d_out, int out_size, void* d_ws, size_t ws_size,
                              hipStream_t stream) {
    (void)n_in; (void)out_size; (void)d_ws; (void)ws_size;
    const float* pcs = (const float*)d_in[0];

    const int N = 4096;
    const int B = in_sizes[0] / (3 * N);   // [B, 3, N]

    float* out_idx = (float*)d_out;                          // [B,N,8,16]
    float* out_msk = out_idx + (size_t)B * N * 8 * KMAX;     // [B,N,8,16]

    const int tiles = B * (N / 16);
    dim3 grid(tiles / WAVES_PER_BLOCK);
    dim3 block(32 * WAVES_PER_BLOCK);
    OctantQuery_wmma_kernel<<<grid, block, 0, stream>>>(pcs, out_idx, out_msk, N);
}